// GCN_80410377716426
// MI455X (gfx1250) — compile-verified
//
#include <hip/hip_runtime.h>

typedef __attribute__((ext_vector_type(2))) float v2f;
typedef __attribute__((ext_vector_type(8))) float v8f;

#define BB 4
#define NN 4096
#define FF 64

// ---- deg init ------------------------------------------------------------
__global__ void k_zero(float* __restrict__ deg) {
    deg[blockIdx.x * 256 + threadIdx.x] = 0.0f;
}

// ---- column sums of A (partial, 64 rows per block) -----------------------
__global__ void k_colsum(const float* __restrict__ A, float* __restrict__ deg) {
    const int j  = blockIdx.x * 256 + threadIdx.x;   // column
    const int b  = blockIdx.z;
    const int r0 = blockIdx.y * 64;                  // row chunk
    const float* Ab = A + (size_t)b * NN * NN;
    float s = 0.0f;
#pragma unroll 4
    for (int r = 0; r < 64; ++r)
        s += Ab[(size_t)(r0 + r) * NN + j];
    atomicAdd(&deg[b * NN + j], s);
}

// ---- d = rsqrt(colsum + 1)  (+1 = self-loop contribution) ----------------
__global__ void k_rsqrt(float* __restrict__ deg) {
    const int i = blockIdx.x * 256 + threadIdx.x;
    deg[i] = rsqrtf(deg[i] + 1.0f);
}

// ---- HWsT[b,o,m] = d[b,m] * sum_f H[b,m,f] * W[f,o]  (TRANSPOSED store) --
// Transposed layout makes the GEMM's B operand a contiguous per-lane float2,
// same access shape as the A operand.
__global__ void k_hw(const float* __restrict__ H, const float* __restrict__ W,
                     const float* __restrict__ d, float* __restrict__ HWsT) {
    __shared__ float  sh[256];    // 4 H rows of 64
    __shared__ float4 tile[64];   // [o] -> 4 consecutive m values
    const int t = threadIdx.x;
    const int rowBase = blockIdx.x * 4;       // 4 (b,m) rows per block
    sh[t] = H[(size_t)rowBase * FF + t];
    __syncthreads();
    const int lr  = t >> 6;                   // 0..3 local row
    const int o   = t & 63;                   // output feature
    const int row = rowBase + lr;             // flat row in [0, B*N)
    float s = 0.0f;
#pragma unroll
    for (int f = 0; f < FF; ++f)
        s += sh[lr * FF + f] * W[f * FF + o];
    ((float*)&tile[o])[lr] = d[row] * s;
    __syncthreads();
    if (t < 64) {                             // 16B store of 4 m's for col t
        const int b  = rowBase / NN;
        const int m0 = rowBase % NN;          // rowBase%4==0 -> same batch
        *(float4*)&HWsT[((size_t)b * FF + t) * NN + m0] = tile[t];
    }
}

// ---- main: out[b,n,o] = d[b,n] * ( (A[b] @ HW[b])[n,o] + HW[b,n,o] ) -----
// 8 waves/block: 2 row-tiles x 4 col-tiles of 16x16, K streamed in steps of 4
// via V_WMMA_F32_16X16X4_F32. Both operands are contiguous per-lane b64 loads;
// A streams from HBM once per workgroup, B (HWsT) is L2-resident (4 MB).
__global__ void __launch_bounds__(256)
k_gcn_main(const float* __restrict__ A, const float* __restrict__ HWsT,
           const float* __restrict__ d, float* __restrict__ out) {
    const int b    = blockIdx.y;
    const int wave = threadIdx.x >> 5;
    const int lane = threadIdx.x & 31;
    const int half = lane >> 4;               // lane group 0 / 1
    const int lm   = lane & 15;

    const int m0 = blockIdx.x * 32 + (wave & 1) * 16;  // output row tile
    const int n0 = (wave >> 1) * 16;                   // output col tile
    const int col = n0 + lm;

    // A operand: lane holds row (m0+lm); v0/v1 = K = 2*half, 2*half+1
    const float* __restrict__ Ab  = A + ((size_t)b * NN + (m0 + lm)) * NN;
    // B operand: lane holds col; v0/v1 = K = 2*half, 2*half+1 (contiguous!)
    const float* __restrict__ BbT = HWsT + ((size_t)b * FF + col) * NN;

    v8f acc = {};
#pragma unroll 8
    for (int k0 = 0; k0 < NN; k0 += 4) {
        v2f a  = *(const v2f*)(Ab  + k0 + 2 * half);
        v2f bv = *(const v2f*)(BbT + k0 + 2 * half);
        acc = __builtin_amdgcn_wmma_f32_16x16x4_f32(
            /*neg_a=*/false, a, /*neg_b=*/false, bv,
            /*c_mod=*/(short)0, acc, /*reuse_a=*/false, /*reuse_b=*/false);
    }

    // Epilogue: C/D layout -> VGPR i holds row m0 + i + 8*half, col = n0+lm.
    // Identity term HW[b,row,col] = HWsT[col][row]: 8 consecutive rows -> b128s.
    const float* __restrict__ db = d + b * NN;
    float* __restrict__ ob = out + (size_t)b * NN * FF;
    const int rbase = m0 + 8 * half;
#pragma unroll
    for (int i = 0; i < 8; ++i) {
        const int row = rbase + i;
        ob[(size_t)row * FF + col] = db[row] * (acc[i] + BbT[row]);
    }
}

extern "C" void kernel_launch(void* const* d_in, const int* in_sizes, int n_in,
                              void* d_out, int out_size, void* d_ws, size_t ws_size,
                              hipStream_t stream) {
    const float* A = (const float*)d_in[0];   // [B,N,N]
    const float* H = (const float*)d_in[1];   // [B,N,F]
    const float* W = (const float*)d_in[2];   // [F,F]
    float* out = (float*)d_out;               // [B,N,F]

    float* deg  = (float*)d_ws;               // B*N floats -> becomes d
    float* HWsT = deg + BB * NN;              // B*F*N floats (transposed HW)

    k_zero<<<BB * NN / 256, 256, 0, stream>>>(deg);

    dim3 gc(NN / 256, 64, BB);
    k_colsum<<<gc, 256, 0, stream>>>(A, deg);

    k_rsqrt<<<BB * NN / 256, 256, 0, stream>>>(deg);

    k_hw<<<BB * NN / 4, 256, 0, stream>>>(H, W, deg, HWsT);

    dim3 gm(NN / 32, BB);
    k_gcn_main<<<gm, 256, 0, stream>>>(A, HWsT, deg, out);
}